// BrainInspiredRNN_75239237091520
// MI455X (gfx1250) — compile-verified
//
#include <hip/hip_runtime.h>

#define B_   4096
#define T_   512
#define NIN_ 3
#define H_   32
#define R_   6
#define NOUT_ 2

typedef float v2f __attribute__((ext_vector_type(2)));
typedef float v8f __attribute__((ext_vector_type(8)));

__device__ __forceinline__ v2f mk2(float a, float b) { v2f r; r.x = a; r.y = b; return r; }
__device__ __forceinline__ v8f zero8() {
  v8f r;
#pragma unroll
  for (int i = 0; i < 8; ++i) r[i] = 0.0f;
  return r;
}

// D = A(16x4 f32) * B(4x16 f32) + C(16x16 f32), wave32 CDNA5 WMMA
__device__ __forceinline__ v8f wmma4(v2f a, v2f b, v8f c) {
  return __builtin_amdgcn_wmma_f32_16x16x4_f32(false, a, false, b, (short)0, c, false, false);
}

__device__ __forceinline__ float fsig(float x) {
  return __builtin_amdgcn_rcpf(1.0f + __expf(-x));
}
__device__ __forceinline__ float ftanh(float x) {
  x = fminf(15.0f, fmaxf(-15.0f, x));
  float e = __expf(-2.0f * x);
  return (1.0f - e) * __builtin_amdgcn_rcpf(1.0f + e);
}

#define LSTRIDE 36  // padded row stride (8B-aligned rows, conflict-free b64 loads)

__global__ __launch_bounds__(32, 1) void lowrank_gru_kernel(
    const float* __restrict__ x, const float* __restrict__ h0,
    const float* __restrict__ Wir, const float* __restrict__ b_ir,
    const float* __restrict__ Wiz, const float* __restrict__ b_iz,
    const float* __restrict__ Win, const float* __restrict__ b_in,
    const float* __restrict__ Ur, const float* __restrict__ Vr, const float* __restrict__ b_hr,
    const float* __restrict__ Uz, const float* __restrict__ Vz, const float* __restrict__ b_hz,
    const float* __restrict__ Un, const float* __restrict__ Vn, const float* __restrict__ b_hn,
    const float* __restrict__ Wout, const float* __restrict__ b_out,
    float* __restrict__ y) {
  __shared__ float lds[16 * LSTRIDE];

  const int lane = threadIdx.x;        // wave32
  const int bl   = lane & 15;          // batch-in-tile (N index, also A-row M)
  const int hf   = lane >> 4;          // lane half selects k pair (4c + 2*hf + {0,1}) per WMMA layout
  const int b0   = blockIdx.x * 16;
  const int b    = b0 + bl;

  // ---- Build constant A fragments (W' rows, extended K = [h(32), 1, x(3)]) ----
  // A layout (16x4 f32): lane<16 holds (M=lane, K=k0..k0+1), lane>=16 (M=lane-16, K=k0+2..k0+3)
  v2f Ar[2][9], Az[2][9], An[2][9], Ax[2], Ay[9];
#pragma unroll
  for (int jb = 0; jb < 2; ++jb) {
    const int j = jb * 16 + bl;
#pragma unroll
    for (int c = 0; c < 8; ++c) {
      const int k0 = 4 * c + 2 * hf;
      float wr0 = 0, wr1 = 0, wz0 = 0, wz1 = 0, wn0 = 0, wn1 = 0;
#pragma unroll
      for (int r = 0; r < R_; ++r) {
        const float ur = Ur[j * R_ + r], uz = Uz[j * R_ + r], un = Un[j * R_ + r];
        wr0 += ur * Vr[k0 * R_ + r];       wr1 += ur * Vr[(k0 + 1) * R_ + r];
        wz0 += uz * Vz[k0 * R_ + r];       wz1 += uz * Vz[(k0 + 1) * R_ + r];
        wn0 += un * Vn[k0 * R_ + r];       wn1 += un * Vn[(k0 + 1) * R_ + r];
      }
      Ar[jb][c] = mk2(wr0, wr1);
      Az[jb][c] = mk2(wz0, wz1);
      An[jb][c] = mk2(wn0, wn1);
    }
    // chunk 8: K slots {32:"1", 33..35: x0..x2}
    if (hf == 0) {
      Ar[jb][8] = mk2(b_ir[j] + b_hr[j], Wir[j * 3 + 0]);
      Az[jb][8] = mk2(b_iz[j] + b_hz[j], Wiz[j * 3 + 0]);
      An[jb][8] = mk2(b_hn[j], 0.0f);            // n-gate recurrent part: + b_hn only
      Ax[jb]    = mk2(b_in[j], Win[j * 3 + 0]);  // n-gate input part: xn + b_in
    } else {
      Ar[jb][8] = mk2(Wir[j * 3 + 1], Wir[j * 3 + 2]);
      Az[jb][8] = mk2(Wiz[j * 3 + 1], Wiz[j * 3 + 2]);
      An[jb][8] = mk2(0.0f, 0.0f);
      Ax[jb]    = mk2(Win[j * 3 + 1], Win[j * 3 + 2]);
    }
  }
  // readout block: rows 0..1 = Wout, rest zero; bias via the "1" slot
#pragma unroll
  for (int c = 0; c < 8; ++c) {
    const int k0 = 4 * c + 2 * hf;
    Ay[c] = (bl < NOUT_) ? mk2(Wout[bl * H_ + k0], Wout[bl * H_ + k0 + 1]) : mk2(0.0f, 0.0f);
  }
  Ay[8] = (bl < NOUT_ && hf == 0) ? mk2(b_out[bl], 0.0f) : mk2(0.0f, 0.0f);

  // ---- Init hidden state: registers in D-layout + LDS copy ----
  // D layout: VGPR i -> (j = jb*16 + 8*hf + i, batch = bl)
  float hD[2][8];
#pragma unroll
  for (int jb = 0; jb < 2; ++jb)
#pragma unroll
    for (int i = 0; i < 8; ++i) {
      const int j = jb * 16 + 8 * hf + i;
      hD[jb][i] = h0[(size_t)b * H_ + j];
      lds[bl * LSTRIDE + j] = hD[jb][i];
    }

  // ---- Recurrence over T ----
#pragma unroll 1
  for (int t = 0; t < T_; ++t) {
    __syncthreads();  // single-wave WG: lowers to S_NOP; orders LDS store->load

    // B fragments of h^(t): lane holds (k = 4c + 2*hf + {0,1}, batch = bl)
    v2f Bh[8];
#pragma unroll
    for (int c = 0; c < 8; ++c)
      Bh[c] = *(const v2f*)&lds[bl * LSTRIDE + 4 * c + 2 * hf];

    const float* xp = x + ((size_t)b * T_ + t) * NIN_;
    const float x0 = xp[0], x1 = xp[1], x2 = xp[2];
    const v2f Bx = hf ? mk2(x1, x2) : mk2(1.0f, x0);

    // readout of h^(t) -> y[:, t-1]
    v8f aY = zero8();
#pragma unroll
    for (int c = 0; c < 9; ++c) aY = wmma4(Ay[c], (c < 8) ? Bh[c] : Bx, aY);
    if (t > 0 && lane < 16) {
      float2 yv = make_float2(aY[0], aY[1]);
      *(float2*)&y[((size_t)b * T_ + (t - 1)) * NOUT_] = yv;
    }

    // gate pre-activations via WMMA accumulation chains
    v8f aR[2], aZ[2], aN[2], aX[2];
#pragma unroll
    for (int jb = 0; jb < 2; ++jb) {
      aR[jb] = zero8(); aZ[jb] = zero8(); aN[jb] = zero8();
#pragma unroll
      for (int c = 0; c < 9; ++c) {
        const v2f Bc = (c < 8) ? Bh[c] : Bx;
        aR[jb] = wmma4(Ar[jb][c], Bc, aR[jb]);
        aZ[jb] = wmma4(Az[jb][c], Bc, aZ[jb]);
        aN[jb] = wmma4(An[jb][c], Bc, aN[jb]);
      }
      aX[jb] = wmma4(Ax[jb], Bx, zero8());
    }

    // elementwise GRU update (positions match hD layout exactly)
#pragma unroll
    for (int jb = 0; jb < 2; ++jb)
#pragma unroll
      for (int i = 0; i < 8; ++i) {
        const float r = fsig(aR[jb][i]);
        const float z = fsig(aZ[jb][i]);
        const float n = ftanh(aX[jb][i] + r * aN[jb][i]);
        hD[jb][i] = (1.0f - z) * n + z * hD[jb][i];
      }

    __syncthreads();
#pragma unroll
    for (int jb = 0; jb < 2; ++jb)
#pragma unroll
      for (int i = 0; i < 8; ++i)
        lds[bl * LSTRIDE + jb * 16 + 8 * hf + i] = hD[jb][i];
  }

  // ---- Epilogue: y[:, T-1] from final hidden state ----
  __syncthreads();
  v2f Bh[8];
#pragma unroll
  for (int c = 0; c < 8; ++c)
    Bh[c] = *(const v2f*)&lds[bl * LSTRIDE + 4 * c + 2 * hf];
  const v2f Bx = hf ? mk2(0.0f, 0.0f) : mk2(1.0f, 0.0f);
  v8f aY = zero8();
#pragma unroll
  for (int c = 0; c < 9; ++c) aY = wmma4(Ay[c], (c < 8) ? Bh[c] : Bx, aY);
  if (lane < 16) {
    float2 yv = make_float2(aY[0], aY[1]);
    *(float2*)&y[((size_t)b * T_ + (T_ - 1)) * NOUT_] = yv;
  }
}

extern "C" void kernel_launch(void* const* d_in, const int* in_sizes, int n_in,
                              void* d_out, int out_size, void* d_ws, size_t ws_size,
                              hipStream_t stream) {
  (void)in_sizes; (void)n_in; (void)out_size; (void)d_ws; (void)ws_size;
  const float* x    = (const float*)d_in[0];
  const float* h0   = (const float*)d_in[1];
  const float* Wir  = (const float*)d_in[2];
  const float* b_ir = (const float*)d_in[3];
  const float* Wiz  = (const float*)d_in[4];
  const float* b_iz = (const float*)d_in[5];
  const float* Win  = (const float*)d_in[6];
  const float* b_in = (const float*)d_in[7];
  const float* Ur   = (const float*)d_in[8];
  const float* Vr   = (const float*)d_in[9];
  const float* b_hr = (const float*)d_in[10];
  const float* Uz   = (const float*)d_in[11];
  const float* Vz   = (const float*)d_in[12];
  const float* b_hz = (const float*)d_in[13];
  const float* Un   = (const float*)d_in[14];
  const float* Vn   = (const float*)d_in[15];
  const float* b_hn = (const float*)d_in[16];
  const float* Wout = (const float*)d_in[17];
  const float* b_out= (const float*)d_in[18];
  float* y = (float*)d_out;

  dim3 grid(B_ / 16);
  dim3 block(32);
  lowrank_gru_kernel<<<grid, block, 0, stream>>>(
      x, h0, Wir, b_ir, Wiz, b_iz, Win, b_in,
      Ur, Vr, b_hr, Uz, Vz, b_hz, Un, Vn, b_hn, Wout, b_out, y);
}